// SoftRank_63101659513440
// MI455X (gfx1250) — compile-verified
//
#include <hip/hip_runtime.h>

// SoftRank: out[b,j,k] = (1/n) * sum_i sigmoid(ALPHA * (x[b,j,k] - x[b,i,k]))
// B=16, N=1024, D=16, ALPHA=1000.
//
// Strategy (MI455X / gfx1250):
//  * sigmoid(z) = 0.5 + 0.5*tanh(z/2)  -> single hardware v_tanh_f32 (TRANS)
//    per element instead of exp+rcp; constant 0.5*n folded out of the loop.
//  * pairwise diff tiles via V_WMMA_F32_16X16X4_F32 (rank-2 product):
//      A (16x4) = [ 500*x_j , -1 , 0 , 0 ]   (K=0,1 in lanes 0-15; K=2,3 zeroed)
//      B (4x16) = [ 1 ; 500*x_i ; * ; * ]    (K=2,3 garbage multiplied by A's zeros)
//      D[m,n]   = 500*(x_j[m] - x_i[n])
//  * whole batch x[b,:,:] staged scaled into LDS in [k][i] layout -> the hot
//    loop's only memory op is one conflict-free ds_load_b32 per 16x16 tile.

typedef __attribute__((ext_vector_type(2))) float v2f;
typedef __attribute__((ext_vector_type(8))) float v8f;

#define SR_B 16
#define SR_N 1024
#define SR_D 16
#define SR_ALPHA_HALF 500.0f   // ALPHA / 2 folded into staging
#define SR_THREADS 512         // 16 wave32's: one wave per feature k

__device__ __forceinline__ float sr_tanh(float x) {
#if __has_builtin(__builtin_amdgcn_tanhf)
  return __builtin_amdgcn_tanhf(x);
#else
  float r;
  asm volatile("v_tanh_f32 %0, %1" : "=v"(r) : "v"(x));
  return r;
#endif
}

__global__ __launch_bounds__(SR_THREADS) void softrank_wmma_kernel(
    const float* __restrict__ x, float* __restrict__ out) {
  // [k][i] layout: hot-loop reads lds[k*1024 + i0 + lane] -> consecutive banks.
  __shared__ float lds[SR_D * SR_N];  // 64 KB

  const int b   = (int)blockIdx.x >> 6;   // 64 j-blocks per batch
  const int jb  = (int)blockIdx.x & 63;
  const int tid = (int)threadIdx.x;

  // ---- Stage x[b,:,:] * (ALPHA/2) into LDS (coalesced global reads) ----
  const float* __restrict__ xb = x + b * (SR_N * SR_D);
#pragma unroll
  for (int t = 0; t < (SR_N * SR_D) / SR_THREADS; ++t) {
    const int idx = tid + t * SR_THREADS;      // idx = i*16 + k
    const int i = idx >> 4;
    const int k = idx & 15;
    lds[(k << 10) + i] = SR_ALPHA_HALF * xb[idx];
  }
  __syncthreads();

  const int  k    = tid >> 5;        // wave id == feature index
  const int  lane = tid & 31;
  const int  ln16 = lane & 15;
  const bool lo   = lane < 16;
  const int  j0   = jb << 4;
  const float* __restrict__ ldsk = &lds[k << 10];

  // A matrix 16x4 f32 (loop-invariant):
  //   VGPR0: lanes 0-15 = K=0 column (500*x_j), lanes 16-31 = K=2 column (0)
  //   VGPR1: lanes 0-15 = K=1 column (-1),      lanes 16-31 = K=3 column (0)
  v2f A;
  A.x = lo ? ldsk[j0 + ln16] : 0.0f;
  A.y = lo ? -1.0f : 0.0f;

  const v8f czero = {0.f, 0.f, 0.f, 0.f, 0.f, 0.f, 0.f, 0.f};
  v8f acc = czero;

  // ---- Main loop: 64 i-tiles; 1 ds_load + 1 wmma + 8 tanh + 8 add each ----
  for (int i0 = 0; i0 < SR_N; i0 += 16) {
    // B matrix 4x16 f32:
    //   VGPR0: lanes 0-15 = K=0 row (ones); lanes 16-31 = K=2 row (ignored)
    //   VGPR1: lanes 0-15 = K=1 row (500*x_i); lanes 16-31 = K=3 row (ignored)
    // K=2,3 contributions vanish because A's K=2,3 columns are zero.
    v2f Bm;
    Bm.x = 1.0f;
    Bm.y = ldsk[i0 + ln16];

    // z[m,n] = 500*(x[j0+m] - x[i0+n])
    v8f z = __builtin_amdgcn_wmma_f32_16x16x4_f32(
        /*neg_a=*/false, A, /*neg_b=*/false, Bm,
        /*c_mod=*/(short)0, czero, /*reuse_a=*/false, /*reuse_b=*/false);

#pragma unroll
    for (int r = 0; r < 8; ++r) acc[r] += sr_tanh(z[r]);
  }

  // ---- Reduce over N (the 16 lanes sharing each output row M) ----
#pragma unroll
  for (int r = 0; r < 8; ++r) {
    float v = acc[r];
    v += __shfl_xor(v, 1, 32);
    v += __shfl_xor(v, 2, 32);
    v += __shfl_xor(v, 4, 32);
    v += __shfl_xor(v, 8, 32);
    acc[r] = v;
  }

  // C/D layout: VGPR r holds M=r (lanes 0-15) and M=r+8 (lanes 16-31).
  // out = (0.5*n + 0.5*sum_tanh)/n = 0.5 + sum/(2n)
  if (ln16 == 0) {  // lanes 0 and 16 write rows 0-7 and 8-15
    const int mbase = lo ? 0 : 8;
    float* __restrict__ ob =
        out + b * (SR_N * SR_D) + (j0 + mbase) * SR_D + k;
#pragma unroll
    for (int r = 0; r < 8; ++r)
      ob[r * SR_D] = 0.5f + acc[r] * (1.0f / (2.0f * SR_N));
  }
}

extern "C" void kernel_launch(void* const* d_in, const int* in_sizes, int n_in,
                              void* d_out, int out_size, void* d_ws, size_t ws_size,
                              hipStream_t stream) {
  (void)in_sizes; (void)n_in; (void)out_size; (void)d_ws; (void)ws_size;
  const float* x = (const float*)d_in[0];
  float* out = (float*)d_out;
  softrank_wmma_kernel<<<dim3(SR_B * (SR_N / 16)), dim3(SR_THREADS), 0, stream>>>(x, out);
}